// TemporalOperator_87806311400070
// MI455X (gfx1250) — compile-verified
//
#include <hip/hip_runtime.h>

typedef __attribute__((ext_vector_type(2))) float v2f;
typedef __attribute__((ext_vector_type(8))) float v8f;
typedef __attribute__((ext_vector_type(4))) int   v4i;

typedef __attribute__((address_space(1))) v4i gv4i;  // global int4
typedef __attribute__((address_space(3))) v4i sv4i;  // LDS int4

namespace {
constexpr int kT     = 2048;  // sequence length
constexpr int kWin   = 128;   // STEPS: softmax window = last 128 columns
constexpr int kBatch = 8192;
constexpr int kRows  = 16;    // rows per wave (one wave32 per workgroup)
constexpr int kPitch = 132;   // LDS row pitch in floats (16B pad)
}

// out[b] = sum(softmax(scale * x[b, T-128:T]) * x[b, T-128:T])
// One wave32 per 16 rows. Stage tile to LDS with GLOBAL_LOAD_ASYNC_TO_LDS_B128,
// lane-partial softmax sums laid out as a WMMA 16x4 f32 A-matrix, reduced with
// V_WMMA_F32_16X16X4_F32 against an all-ones B:
//   D[r][n] = partial_r + partial_{r+16} = full 128-element row sum.
__global__ __launch_bounds__(32)
void maxish_tail_kernel(const float* __restrict__ x,
                        const int*   __restrict__ scale,
                        float* __restrict__ out)
{
    __shared__ float tile[kRows * kPitch];
    const int   lane    = threadIdx.x;           // 0..31
    const int   rowBase = blockIdx.x * kRows;
    const float s       = (float)scale[0];

    // ---- Stage: global -> LDS, fully coalesced ----
    // Iteration i: 32 lanes x 16B = one full 512B row tail of row (rowBase+i).
#if __has_builtin(__builtin_amdgcn_global_load_async_to_lds_b128)
#pragma unroll
    for (int i = 0; i < kRows; ++i) {
        const float* g = x + (size_t)(rowBase + i) * kT + (kT - kWin) + lane * 4;
        __builtin_amdgcn_global_load_async_to_lds_b128(
            (gv4i*)g,
            (sv4i*)&tile[i * kPitch + lane * 4],
            /*imm offset*/ 0, /*cpol*/ 0);
    }
    asm volatile("s_wait_asynccnt 0" ::: "memory");  // ASYNCcnt -> 0: data is in LDS
#else
#pragma unroll
    for (int i = 0; i < kRows; ++i) {
        const float4 v = *reinterpret_cast<const float4*>(
            x + (size_t)(rowBase + i) * kT + (kT - kWin) + lane * 4);
        *reinterpret_cast<float4*>(&tile[i * kPitch + lane * 4]) = v;
    }
#endif
    __syncthreads();  // single-wave WG: degenerates to counter waits

    // Row partition: lane (ri, h) owns the contiguous half-row [64h, 64h+63].
    // Lanes ri and ri+16 together cover row ri, matching the WMMA 16x4 f32
    // A layout (lane r -> A[r][0..1], lane r+16 -> A[r][2..3]).
    const int     ri   = lane & 15;
    const int     h    = lane >> 4;
    const float4* rowp = reinterpret_cast<const float4*>(&tile[ri * kPitch + 64 * h]);

    // ---- Pass 1: per-row max (softmax stabilization, matches jax.nn.softmax) ----
    float m = -__builtin_inff();
#pragma unroll
    for (int c = 0; c < 16; ++c) {
        const float4 w = rowp[c];
        m = fmaxf(m, fmaxf(fmaxf(w.x, w.y), fmaxf(w.z, w.w)));
    }
    m = fmaxf(m, __shfl_xor(m, 16));  // combine the two half-row owners

    // ---- Pass 2: lane-partial  sum(e)  and  sum(e*w) ----
    float se = 0.f, sew = 0.f;
#pragma unroll
    for (int c = 0; c < 16; ++c) {
        const float4 w  = rowp[c];
        const float  e0 = __expf(s * (w.x - m));
        const float  e1 = __expf(s * (w.y - m));
        const float  e2 = __expf(s * (w.z - m));
        const float  e3 = __expf(s * (w.w - m));
        se  += (e0 + e1) + (e2 + e3);
        sew += (e0 * w.x + e1 * w.y) + (e2 * w.z + e3 * w.w);
    }

    // ---- WMMA reduction: A = lane partials (K slots 1 and 3 zero), B = ones.
    v2f a_se  = { se,  0.f };
    v2f a_sew = { sew, 0.f };
    v2f ones  = { 1.f, 1.f };
    v8f zero  = {};
    v8f d_se  = __builtin_amdgcn_wmma_f32_16x16x4_f32(
        false, a_se,  false, ones, (short)0, zero, false, false);
    v8f d_sew = __builtin_amdgcn_wmma_f32_16x16x4_f32(
        false, a_sew, false, ones, (short)0, zero, false, false);

    // D VGPR v: lanes 0-15 -> row v (N=lane), lanes 16-31 -> row v+8.
    // d_se >= 1 always (window max contributes e^0 = 1), so v_rcp_f32 is safe;
    // avoids the full IEEE division expansion (div_scale/div_fmas/div_fixup).
    if (ri == 0) {
        float* o = out + rowBase + h * 8;
#pragma unroll
        for (int v = 0; v < 8; ++v) o[v] = d_sew[v] * __builtin_amdgcn_rcpf(d_se[v]);
    }
}

extern "C" void kernel_launch(void* const* d_in, const int* in_sizes, int n_in,
                              void* d_out, int out_size, void* d_ws, size_t ws_size,
                              hipStream_t stream)
{
    const float* x     = (const float*)d_in[0];
    // d_in[1] (h0) is provably dead: with T=2048 >= 128, the init window is
    // fully flushed before the final (and only returned) step.
    const int*   scale = (const int*)d_in[2];
    float*       out   = (float*)d_out;

    const int nBlocks = kBatch / kRows;  // 512 single-wave workgroups
    maxish_tail_kernel<<<dim3(nBlocks), dim3(32), 0, stream>>>(x, scale, out);
}